// CustomMonteCarloSilhouetteRaysampler_81844896793054
// MI455X (gfx1250) — compile-verified
//
#include <hip/hip_runtime.h>
#include <hip/hip_bf16.h>
#include <stdint.h>

// ---------------- static config (mirrors reference) ----------------
#define BB      32
#define HH      512
#define WW      512
#define HWPIX   (HH * WW)        // 262144
#define N_RAYS  4096
#define N_IN    3072
#define N_OUT   1024
#define N_PTS   128
#define MIN_D   0.1f
#define MAX_D   10.0f
#define NTILES  1024             // tiles of 256 pixels per image
#define TILEPIX 256

// ---------------- threefry2x32 (JAX-compatible) ----------------
__device__ __forceinline__ uint32_t rotl32(uint32_t x, int r) {
    return (x << r) | (x >> (32 - r));
}

__device__ __forceinline__ void tf2x32(uint32_t k0, uint32_t k1,
                                       uint32_t c0, uint32_t c1,
                                       uint32_t& o0, uint32_t& o1) {
    uint32_t ks[3];
    ks[0] = k0; ks[1] = k1; ks[2] = k0 ^ k1 ^ 0x1BD11BDAu;
    uint32_t x0 = c0 + ks[0];
    uint32_t x1 = c1 + ks[1];
    const int RA[4] = {13, 15, 26, 6};
    const int RB[4] = {17, 29, 16, 24};
#pragma unroll
    for (int i = 0; i < 5; ++i) {
        const int* r = (i & 1) ? RB : RA;
#pragma unroll
        for (int j = 0; j < 4; ++j) {
            x0 += x1;
            x1 = rotl32(x1, r[j]);
            x1 ^= x0;
        }
        x0 += ks[(i + 1) % 3];
        x1 += ks[(i + 2) % 3] + (uint32_t)(i + 1);
    }
    o0 = x0; o1 = x1;
}

__device__ __forceinline__ float bits_to_u01(uint32_t bits) {
    uint32_t f = (bits >> 9) | 0x3f800000u;
    return __uint_as_float(f) - 1.0f;
}

// ---------------- kernel 1: per-batch tile counts + scan ----------------
// prefix_*: [B][NTILES] exclusive prefix of selected-pixel counts per 256-px tile
__global__ void cdf_kernel(const float* __restrict__ masks,
                           int* __restrict__ prefix_in, int* __restrict__ prefix_out,
                           int* __restrict__ total_in, int* __restrict__ total_out) {
    const int b = blockIdx.x;
    const int t = threadIdx.x;          // 0..255
    const float* m = masks + (size_t)b * HWPIX;

    int cin_l[4], cout_l[4];
    int tin = 0, tout = 0;
#pragma unroll
    for (int q = 0; q < 4; ++q) {
        const int tile = t * 4 + q;     // contiguous tiles per thread
        const float4* m4 = (const float4*)(m + tile * TILEPIX);
        int ci = 0, co = 0;
        for (int p = 0; p < TILEPIX / 4; ++p) {
            float4 v = m4[p];
            ci += ((v.x > 0.5f) ? 1 : 0) + ((v.y > 0.5f) ? 1 : 0) +
                  ((v.z > 0.5f) ? 1 : 0) + ((v.w > 0.5f) ? 1 : 0);
            co += ((v.x < 0.5f) ? 1 : 0) + ((v.y < 0.5f) ? 1 : 0) +
                  ((v.z < 0.5f) ? 1 : 0) + ((v.w < 0.5f) ? 1 : 0);
        }
        cin_l[q] = ci; cout_l[q] = co;
        tin += ci; tout += co;
    }

    __shared__ int si[256], so[256];
    si[t] = tin; so[t] = tout;
    __syncthreads();
    // Hillis-Steele inclusive scan over 256 thread sums
    for (int off = 1; off < 256; off <<= 1) {
        int vi = (t >= off) ? si[t - off] : 0;
        int vo = (t >= off) ? so[t - off] : 0;
        __syncthreads();
        si[t] += vi; so[t] += vo;
        __syncthreads();
    }
    int run_i = si[t] - tin;   // exclusive prefix for this thread's first tile
    int run_o = so[t] - tout;
#pragma unroll
    for (int q = 0; q < 4; ++q) {
        prefix_in [b * NTILES + t * 4 + q] = run_i;
        prefix_out[b * NTILES + t * 4 + q] = run_o;
        run_i += cin_l[q];
        run_o += cout_l[q];
    }
    if (t == 255) { total_in[b] = run_i; total_out[b] = run_o; }
}

// ---------------- kernel 2: sample rays + unproject ----------------
__global__ void sample_kernel(const float* __restrict__ masks,
                              const float* __restrict__ Rm, const float* __restrict__ Tm,
                              const float* __restrict__ focal, const float* __restrict__ principal,
                              const int* __restrict__ prefix_in, const int* __restrict__ prefix_out,
                              const int* __restrict__ total_in, const int* __restrict__ total_out,
                              float* __restrict__ origins, float* __restrict__ dirs_o,
                              float* __restrict__ xy_o, float* __restrict__ norms) {
    const int gid = blockIdx.x * blockDim.x + threadIdx.x;   // exact grid: B*N_RAYS
    const int b = gid / N_RAYS;
    const int s = gid % N_RAYS;
    const bool inside = (s < N_IN);
    const int n  = inside ? N_IN : N_OUT;
    const int si = inside ? s : (s - N_IN);

    // master key: jax.random.key(1) for inside, key(2) for outside -> (0, seed)
    const uint32_t mk0 = 0u, mk1 = inside ? 1u : 2u;

    // split(master, 32) -> batch key b  (flat reshape semantics of threefry_split)
    uint32_t bk0, bk1, o0, o1;
    if (b < 16) {
        tf2x32(mk0, mk1, (uint32_t)(2 * b),     (uint32_t)(2 * b + 32), o0, o1); bk0 = o0;
        tf2x32(mk0, mk1, (uint32_t)(2 * b + 1), (uint32_t)(2 * b + 33), o0, o1); bk1 = o0;
    } else {
        tf2x32(mk0, mk1, (uint32_t)(2 * b - 32), (uint32_t)(2 * b),     o0, o1); bk0 = o1;
        tf2x32(mk0, mk1, (uint32_t)(2 * b - 31), (uint32_t)(2 * b + 1), o0, o1); bk1 = o1;
    }

    // split(batch_key, 2) -> k1 (u draw), k2 (jitter)
    uint32_t f0, f1, f2, f3;
    tf2x32(bk0, bk1, 0u, 2u, f0, f2);
    tf2x32(bk0, bk1, 1u, 3u, f1, f3);
    const uint32_t k10 = f0, k11 = f1, k20 = f2, k21 = f3;

    // u = uniform(k1, (n,))[si] * total
    const int half = n >> 1;
    uint32_t ubits;
    if (si < half) { tf2x32(k10, k11, (uint32_t)si,          (uint32_t)(si + half), o0, o1); ubits = o0; }
    else           { tf2x32(k10, k11, (uint32_t)(si - half), (uint32_t)si,          o0, o1); ubits = o1; }
    const int total = inside ? total_in[b] : total_out[b];
    float u = bits_to_u01(ubits) * (float)total;
    int target = (int)floorf(u) + 1;
    if (target > total) target = total;
    if (target < 1) target = 1;

    // binary search: largest tile with exclusive prefix < target
    const int* pref = (inside ? prefix_in : prefix_out) + b * NTILES;
    int lo = 0, hi = NTILES - 1;
    while (lo < hi) {
        int mid = (lo + hi + 1) >> 1;
        if (pref[mid] < target) lo = mid; else hi = mid - 1;
    }
    const int tile = lo;
    int run = pref[tile];
    const float4* m4 = (const float4*)(masks + (size_t)b * HWPIX + tile * TILEPIX);
    int j = tile * TILEPIX;
    bool found = false;
    for (int q = 0; q < TILEPIX / 4 && !found; ++q) {
        float4 v = m4[q];
        float vv[4] = {v.x, v.y, v.z, v.w};
#pragma unroll
        for (int c = 0; c < 4; ++c) {
            bool sel = inside ? (vv[c] > 0.5f) : (vv[c] < 0.5f);
            if (sel) {
                run++;
                if (run == target) { j = tile * TILEPIX + q * 4 + c; found = true; break; }
            }
        }
    }
    const int row = j >> 9;          // / WW
    const int col = j & (WW - 1);

    // jitter = uniform(k2, (n,2))[si]  (flat indices 2*si, 2*si+1; same half always)
    uint32_t jb0, jb1;
    {
        const int e0 = 2 * si, e1 = 2 * si + 1;
        if (e1 < n) {
            tf2x32(k20, k21, (uint32_t)e0, (uint32_t)(e0 + n), o0, o1); jb0 = o0;
            tf2x32(k20, k21, (uint32_t)e1, (uint32_t)(e1 + n), o0, o1); jb1 = o0;
        } else {
            tf2x32(k20, k21, (uint32_t)(e0 - n), (uint32_t)e0, o0, o1); jb0 = o1;
            tf2x32(k20, k21, (uint32_t)(e1 - n), (uint32_t)e1, o0, o1); jb1 = o1;
        }
    }
    const float pr = (float)row + bits_to_u01(jb0);
    const float pc = (float)col + bits_to_u01(jb1);
    const float ynorm = 2.0f * pr / (float)HH - 1.0f;
    const float xnorm = 2.0f * pc / (float)WW - 1.0f;
    // rays_xy = (x=col_norm, y=row_norm)
    const float rx = xnorm, ry = ynorm;

    // ray bundle
    const float ndx = -rx, ndy = -ry;
    const float fx = focal[b * 2 + 0], fy = focal[b * 2 + 1];
    const float px = principal[b * 2 + 0], py = principal[b * 2 + 1];
    const float c1x = (ndx - px) / fx, c1y = (ndy - py) / fy, c1z = 1.0f;
    const float c2x = 2.0f * c1x,      c2y = 2.0f * c1y,      c2z = 2.0f;

    const float* Rb = Rm + b * 9;
    const float* Tb = Tm + b * 3;
    const float v1x = c1x - Tb[0], v1y = c1y - Tb[1], v1z = c1z - Tb[2];
    const float v2x = c2x - Tb[0], v2y = c2y - Tb[1], v2z = c2z - Tb[2];

    float p1[3], p2[3];
#pragma unroll
    for (int k = 0; k < 3; ++k) {
        p1[k] = v1x * Rb[k * 3 + 0] + v1y * Rb[k * 3 + 1] + v1z * Rb[k * 3 + 2];
        p2[k] = v2x * Rb[k * 3 + 0] + v2y * Rb[k * 3 + 1] + v2z * Rb[k * 3 + 2];
    }
    float d0 = p2[0] - p1[0], d1 = p2[1] - p1[1], d2 = p2[2] - p1[2];
    origins[(size_t)gid * 3 + 0] = p1[0] - d0;
    origins[(size_t)gid * 3 + 1] = p1[1] - d1;
    origins[(size_t)gid * 3 + 2] = p1[2] - d2;
    dirs_o [(size_t)gid * 3 + 0] = d0;
    dirs_o [(size_t)gid * 3 + 1] = d1;
    dirs_o [(size_t)gid * 3 + 2] = d2;
    xy_o   [(size_t)gid * 2 + 0] = rx;
    xy_o   [(size_t)gid * 2 + 1] = ry;
    norms[gid] = sqrtf(d0 * d0 + d1 * d1 + d2 * d2);
}

// ---------------- kernel 3: lengths = depths (x) norms via WMMA ----------------
// Each wave: 16 rays x 128 depths via 8x V_WMMA_F32_16X16X4_F32 outer products.
// D tiles are staged in LDS (tile is contiguous in the output), then drained to
// global with the gfx1250 async LDS->global DMA path (512B per instruction,
// tracked on ASYNCcnt) instead of scattered b32 stores.
typedef __attribute__((ext_vector_type(2))) float v2f;
typedef __attribute__((ext_vector_type(8))) float v8f;

#define LW_WAVES 4   // waves per block; 4 * 8KB tile = 32KB static LDS

__global__ void lengths_kernel(const float* __restrict__ norms,
                               float* __restrict__ lengths) {
    __shared__ float tb[LW_WAVES][16 * N_PTS];   // 8KB per wave
    const int wib   = threadIdx.x >> 5;          // wave in block
    const int lane  = threadIdx.x & 31;
    const int gwave = blockIdx.x * LW_WAVES + wib;
    const int r0    = gwave * 16;                // grid exactly covers B*N_RAYS

    // A-matrix 16x4 f32: VGPR0 = K=0 (lanes 0-15) / K=2 (lanes 16-31); VGPR1 = K=1 / K=3.
    float nv = 0.0f;
    if (lane < 16) nv = norms[r0 + lane];
    v2f a;
    a.x = (lane < 16) ? nv : 0.0f;
    a.y = 0.0f;

    const float step = (MAX_D - MIN_D) / 127.0f;
    const int rowoff = (lane < 16) ? 0 : 8;
#pragma unroll
    for (int t = 0; t < 8; ++t) {
        const int col = t * 16 + (lane & 15);
        const float depth = MIN_D + (float)col * step;
        // B-matrix 4x16 f32: row K=0 (lanes 0-15) carries 16 depths, zeros elsewhere.
        v2f bm;
        bm.x = (lane < 16) ? depth : 0.0f;
        bm.y = 0.0f;
        v8f c = {};
        c = __builtin_amdgcn_wmma_f32_16x16x4_f32(
                /*neg_a=*/false, a, /*neg_b=*/false, bm,
                /*c_mod=*/(short)0, c, /*reuse_a=*/false, /*reuse_b=*/false);
        // D 16x16 f32: VGPR v -> M = v (lanes 0-15) or v+8 (lanes 16-31), N = lane%16
#pragma unroll
        for (int v = 0; v < 8; ++v) {
            tb[wib][(v + rowoff) * N_PTS + col] = c[v];
        }
    }

    // Wave-private staging: only need this wave's DS stores complete (in-order per wave).
    asm volatile("s_wait_dscnt 0x0" ::: "memory");

    // Drain the contiguous 8KB tile: 16 iters x 32 lanes x 16B = 8192B, b128 async DMA.
    const unsigned long long gbase =
        (unsigned long long)(uintptr_t)lengths + (unsigned long long)r0 * N_PTS * 4ull;
    const uint32_t lbase = (uint32_t)(uintptr_t)&tb[wib][0];
#pragma unroll
    for (int i = 0; i < 16; ++i) {
        const uint32_t off = (uint32_t)(i * 32 + lane) * 16u;   // bytes
        unsigned long long gaddr = gbase + off;
        uint32_t laddr = lbase + off;
        asm volatile("global_store_async_from_lds_b128 %0, %1, off"
                     :: "v"(gaddr), "v"(laddr) : "memory");
    }
    asm volatile("s_wait_asynccnt 0x0" ::: "memory");
}

// ---------------- launch ----------------
extern "C" void kernel_launch(void* const* d_in, const int* in_sizes, int n_in,
                              void* d_out, int out_size, void* d_ws, size_t ws_size,
                              hipStream_t stream) {
    const float* masks     = (const float*)d_in[0];
    const float* Rm        = (const float*)d_in[1];
    const float* Tm        = (const float*)d_in[2];
    const float* focal     = (const float*)d_in[3];
    const float* principal = (const float*)d_in[4];

    float* out     = (float*)d_out;
    float* origins = out;                                       // B*N*3
    float* dirs    = origins + (size_t)BB * N_RAYS * 3;         // B*N*3
    float* lengths = dirs    + (size_t)BB * N_RAYS * 3;         // B*N*128
    float* xy      = lengths + (size_t)BB * N_RAYS * N_PTS;     // B*N*2

    int* prefix_in  = (int*)d_ws;                               // B*NTILES
    int* prefix_out = prefix_in + BB * NTILES;                  // B*NTILES
    int* total_in   = prefix_out + BB * NTILES;                 // B
    int* total_out  = total_in + BB;                            // B
    float* norms    = (float*)(total_out + BB);                 // B*N

    cdf_kernel<<<BB, 256, 0, stream>>>(masks, prefix_in, prefix_out, total_in, total_out);

    const int nthreads = BB * N_RAYS;                           // 131072
    sample_kernel<<<nthreads / 256, 256, 0, stream>>>(
        masks, Rm, Tm, focal, principal,
        prefix_in, prefix_out, total_in, total_out,
        origins, dirs, xy, norms);

    // B*N/16 waves = 8192 waves; LW_WAVES waves per block
    lengths_kernel<<<(BB * N_RAYS / 16) / LW_WAVES, LW_WAVES * 32, 0, stream>>>(norms, lengths);
}